// LnnDynamics_75067438399931
// MI455X (gfx1250) — compile-verified
//
#include <hip/hip_runtime.h>

// LNN dynamics on gfx1250: f32 WMMA (16x16x4) batched-small-GEMM pipeline.
// One wave32 = 16-sample tile. Weights live in LDS in fragment-ready
// (transposed, stride-34) layout; activations staged per-wave in LDS.
// Round 3: raw hardware transcendentals (v_exp_f32 / v_log_f32 / v_rcp_f32
// via amdgcn builtins) -- log argument is in [1,2], exp argument bounded, so
// the libm range/denorm fixup chains are unnecessary.

typedef float v2f __attribute__((ext_vector_type(2)));
typedef float v8f __attribute__((ext_vector_type(8)));

#define WAVES 4
#define BLOCK 128
#define WSTR 34

#define LOG2E 1.4426950408889634f
#define LN2 0.6931471805599453f

// ---- LDS layout (float indices) ----
// per-MLP weight block (order: 0=m, 1=u, 2=d)
#define O_WT0 0                 // 32x8   : WT0[n][k] = W0[k][n] (k<6 else 0)
#define O_WTH 256               // 4*32*34: WTh[l][n][k] = Wh[l][k][n]
#define O_WTO 4608              // 48*34  : WTo[n][k] = Wo[k][n] (n<36 else 0)
#define O_B0 6240
#define O_BH 6272
#define O_BO 6400               // 48 (padded)
#define MLPSZ 6448
#define O_UPW (3 * MLPSZ)       // 4*32*34: u plain Wh (for backward W^T matmuls)
#define O_UW0P (O_UPW + 4352)   // 16*34  : u plain W0 rows padded to 16
#define O_UWO (O_UW0P + 544)    // 32     : u output weight vector
#define O_AM (O_UWO + 32)       // 36 -> pad 40 : A matrix
#define O_WAVE 24320
// per-wave scratch
#define PWSZ 3264
#define PW_XB 0                 // 16x8  x (padded)
#define PW_TB 128               // 16x8  xd (padded)
#define PW_TAU 256              // 16x8  tau (padded)
#define PW_HB 384               // 16x34 activations
#define PW_GB 928               // 16x34 tangent / delta
#define PW_O1 1472              // 16x48 primal out (D or M)
#define PW_O2 2240              // 16x48 tangent out (dM)
#define PW_PB 3008              // 16x16 tau_pot dump
#define SMEM_FLOATS (O_WAVE + WAVES * PWSZ)
#define SMEM_BYTES (SMEM_FLOATS * 4)

__device__ __forceinline__ v8f wmma4(v2f a, v2f b, v8f c) {
  // (neg_a, A, neg_b, B, c_mod, C, reuse_a, reuse_b)
  return __builtin_amdgcn_wmma_f32_16x16x4_f32(false, a, false, b, (short)0, c,
                                               false, false);
}

__device__ __forceinline__ v8f splat8(float x) {
  v8f r;
#pragma unroll
  for (int i = 0; i < 8; ++i) r[i] = x;
  return r;
}

// Intra-wave LDS fence: DS ops are in-order per wave; we just need the wait
// and a compiler scheduling barrier.
__device__ __forceinline__ void wfence() {
  __builtin_amdgcn_wave_barrier();
  asm volatile("s_wait_dscnt 0" ::: "memory");
  __builtin_amdgcn_wave_barrier();
}

// softplus(z) = max(z,0) + ln2 * log2(1 + 2^(-|z|*log2e))
__device__ __forceinline__ float softplusf(float z) {
  float t = __builtin_amdgcn_exp2f(-fabsf(z) * LOG2E);
  return fmaxf(z, 0.f) + LN2 * __builtin_amdgcn_logf(1.f + t);
}
// Fused softplus + sigmoid: shares 2^(-|z|*log2e); sigmoid via v_rcp.
__device__ __forceinline__ void spsig(float z, float& sp, float& sg) {
  float t = __builtin_amdgcn_exp2f(-fabsf(z) * LOG2E);
  float u = 1.f + t;
  sp = fmaxf(z, 0.f) + LN2 * __builtin_amdgcn_logf(u);
  sg = (z >= 0.f ? 1.f : t) * __builtin_amdgcn_rcpf(u);
}

// A fragment (row lr of aB) and B fragment (row nt*16+lr of wB) share the same
// addressing: base + row*stride + 4*kstep + 2*half  (ds_load_b64).
template <int KS, int NT>
__device__ __forceinline__ void mmacc(const float* aB, int aS, const float* wB,
                                      int wS, v8f* acc, int lr, int half) {
#pragma unroll
  for (int s = 0; s < KS; ++s) {
    v2f a = *(const v2f*)(aB + lr * aS + 4 * s + 2 * half);
#pragma unroll
    for (int nt = 0; nt < NT; ++nt) {
      v2f b = *(const v2f*)(wB + (nt * 16 + lr) * wS + 4 * s + 2 * half);
      acc[nt] = wmma4(a, b, acc[nt]);
    }
  }
}

template <int KS, int NT>
__device__ __forceinline__ void mmacc_dual(const float* aB, const float* tB,
                                           int aS, const float* wB, int wS,
                                           v8f* pa, v8f* ta, int lr, int half) {
#pragma unroll
  for (int s = 0; s < KS; ++s) {
    v2f a = *(const v2f*)(aB + lr * aS + 4 * s + 2 * half);
    v2f t = *(const v2f*)(tB + lr * aS + 4 * s + 2 * half);
#pragma unroll
    for (int nt = 0; nt < NT; ++nt) {
      v2f b = *(const v2f*)(wB + (nt * 16 + lr) * wS + 4 * s + 2 * half);
      pa[nt] = wmma4(a, b, pa[nt]);
      ta[nt] = wmma4(t, b, ta[nt]);
    }
  }
}

// C layout: vgpr i, lanes 0-15 -> (m=i, n=lr+16*nt), lanes 16-31 -> m=i+8.
__device__ __forceinline__ void storeC(float* buf, int stride, int nt, v8f v,
                                       int lr, int half) {
#pragma unroll
  for (int i = 0; i < 8; ++i)
    buf[(i + 8 * half) * stride + nt * 16 + lr] = v[i];
}

__global__ void __launch_bounds__(BLOCK) lnn_kernel(
    const float* __restrict__ inp, const float* __restrict__ Am,
    const float* mW0, const float* mb0, const float* mWh, const float* mbh,
    const float* mWo, const float* mbo, const float* uW0, const float* ub0,
    const float* uWh, const float* ubh, const float* uWo, const float* ubo,
    const float* dW0, const float* db0, const float* dWh, const float* dbh,
    const float* dWo, const float* dbo, float* __restrict__ outp, int nTiles) {
  extern __shared__ float smem[];
  const int tid = threadIdx.x;
  const int warp = tid >> 5;
  const int lane = tid & 31;
  const int half = lane >> 4;
  const int lr = lane & 15;

  // ---- preprocess weights into LDS (fragment-ready layouts) ----
  {
    const float* W0s[3] = {mW0, uW0, dW0};
    const float* b0s[3] = {mb0, ub0, db0};
    const float* Whs[3] = {mWh, uWh, dWh};
    const float* bhs[3] = {mbh, ubh, dbh};
    const float* Wos[3] = {mWo, uWo, dWo};
    const float* bos[3] = {mbo, ubo, dbo};
    for (int p = 0; p < 3; ++p) {
      float* Bp = smem + p * MLPSZ;
      for (int i = tid; i < 32 * 8; i += BLOCK) {
        int n = i >> 3, k = i & 7;
        Bp[O_WT0 + n * 8 + k] = (k < 6) ? W0s[p][k * 32 + n] : 0.f;
      }
      for (int i = tid; i < 4 * 32 * 32; i += BLOCK) {
        int l = i >> 10, r = (i >> 5) & 31, c = i & 31;
        Bp[O_WTH + l * 32 * WSTR + c * WSTR + r] = Whs[p][i];
      }
      if (p != 1) {
        for (int i = tid; i < 48 * 32; i += BLOCK) {
          int n = i >> 5, k = i & 31;
          Bp[O_WTO + n * WSTR + k] = (n < 36) ? Wos[p][k * 36 + n] : 0.f;
        }
        for (int i = tid; i < 48; i += BLOCK)
          Bp[O_BO + i] = (i < 36) ? bos[p][i] : 0.f;
      }
      for (int i = tid; i < 32; i += BLOCK) Bp[O_B0 + i] = b0s[p][i];
      for (int i = tid; i < 128; i += BLOCK) Bp[O_BH + i] = bhs[p][i];
    }
    for (int i = tid; i < 4 * 32 * 32; i += BLOCK) {
      int l = i >> 10, r = (i >> 5) & 31, c = i & 31;
      smem[O_UPW + l * 32 * WSTR + r * WSTR + c] = uWh[i];
    }
    for (int i = tid; i < 16 * 32; i += BLOCK) {
      int r = i >> 5, c = i & 31;
      smem[O_UW0P + r * WSTR + c] = (r < 6) ? uW0[r * 32 + c] : 0.f;
    }
    for (int i = tid; i < 32; i += BLOCK) smem[O_UWO + i] = uWo[i];
    for (int i = tid; i < 36; i += BLOCK) smem[O_AM + i] = Am[i];
  }
  __syncthreads();

  float* wv = smem + O_WAVE + warp * PWSZ;
  float* xb = wv + PW_XB;
  float* tb = wv + PW_TB;
  float* taub = wv + PW_TAU;
  float* hb = wv + PW_HB;
  float* gb = wv + PW_GB;
  float* o1 = wv + PW_O1;
  float* o2 = wv + PW_O2;
  float* pb = wv + PW_PB;

  const int waveId = blockIdx.x * WAVES + warp;
  const int waveCnt = gridDim.x * WAVES;

  for (int tile = waveId; tile < nTiles; tile += waveCnt) {
    const int s0 = tile * 16;
    // stage inputs: x | xd | tau, K-padded to 8
    if (half == 0) {
      const float* row = inp + (size_t)(s0 + lr) * 18;
#pragma unroll
      for (int c = 0; c < 6; ++c) {
        xb[lr * 8 + c] = row[c];
        tb[lr * 8 + c] = row[6 + c];
        taub[lr * 8 + c] = row[12 + c];
      }
      xb[lr * 8 + 6] = 0.f;  xb[lr * 8 + 7] = 0.f;
      tb[lr * 8 + 6] = 0.f;  tb[lr * 8 + 7] = 0.f;
      taub[lr * 8 + 6] = 0.f; taub[lr * 8 + 7] = 0.f;
    }
    wfence();

    float td[6] = {0.f, 0.f, 0.f, 0.f, 0.f, 0.f};

    // ===================== D-MLP forward -> tau_d =====================
    {
      const float* Bp = smem + 2 * MLPSZ;
      v8f acc[2];
      acc[0] = splat8(Bp[O_B0 + lr]);
      acc[1] = splat8(Bp[O_B0 + 16 + lr]);
      mmacc<2, 2>(xb, 8, Bp + O_WT0, 8, acc, lr, half);
#pragma unroll
      for (int nt = 0; nt < 2; ++nt) {
        v8f h;
#pragma unroll
        for (int i = 0; i < 8; ++i) h[i] = softplusf(acc[nt][i]);
        storeC(hb, WSTR, nt, h, lr, half);
      }
      wfence();
#pragma unroll
      for (int l = 0; l < 4; ++l) {
        v8f a2[2];
        a2[0] = splat8(Bp[O_BH + l * 32 + lr]);
        a2[1] = splat8(Bp[O_BH + l * 32 + 16 + lr]);
        mmacc<8, 2>(hb, WSTR, Bp + O_WTH + l * 32 * WSTR, WSTR, a2, lr, half);
#pragma unroll
        for (int nt = 0; nt < 2; ++nt) {
          v8f h;
#pragma unroll
          for (int i = 0; i < 8; ++i) h[i] = softplusf(a2[nt][i]);
          storeC(hb, WSTR, nt, h, lr, half);
        }
        wfence();
      }
      v8f o[3];
#pragma unroll
      for (int nt = 0; nt < 3; ++nt) o[nt] = splat8(Bp[O_BO + nt * 16 + lr]);
      mmacc<8, 3>(hb, WSTR, Bp + O_WTO, WSTR, o, lr, half);
#pragma unroll
      for (int nt = 0; nt < 3; ++nt) storeC(o1, 48, nt, o[nt], lr, half);
      wfence();
      if (half == 0) {
#pragma unroll
        for (int i = 0; i < 6; ++i) {
          float s = 0.f;
#pragma unroll
          for (int j = 0; j < 6; ++j) s += o1[lr * 48 + i * 6 + j] * tb[lr * 8 + j];
          td[i] = s;
        }
      }
    }

    // ============ U-MLP forward + backward -> tau_pot (in pb) ============
    {
      const float* Bp = smem + 1 * MLPSZ;
      v8f sg[5][2];  // sigmoid(z_l) in C layout, kept in registers
      {
        v8f acc[2];
        acc[0] = splat8(Bp[O_B0 + lr]);
        acc[1] = splat8(Bp[O_B0 + 16 + lr]);
        mmacc<2, 2>(xb, 8, Bp + O_WT0, 8, acc, lr, half);
#pragma unroll
        for (int nt = 0; nt < 2; ++nt) {
          v8f h;
#pragma unroll
          for (int i = 0; i < 8; ++i) {
            float sp, sgv;
            spsig(acc[nt][i], sp, sgv);
            h[i] = sp;
            sg[0][nt][i] = sgv;
          }
          storeC(hb, WSTR, nt, h, lr, half);
        }
        wfence();
      }
#pragma unroll
      for (int l = 0; l < 4; ++l) {
        v8f acc[2];
        acc[0] = splat8(Bp[O_BH + l * 32 + lr]);
        acc[1] = splat8(Bp[O_BH + l * 32 + 16 + lr]);
        mmacc<8, 2>(hb, WSTR, Bp + O_WTH + l * 32 * WSTR, WSTR, acc, lr, half);
#pragma unroll
        for (int nt = 0; nt < 2; ++nt) {
          v8f h;
#pragma unroll
          for (int i = 0; i < 8; ++i) {
            float sp, sgv;
            spsig(acc[nt][i], sp, sgv);
            h[i] = sp;
            sg[l + 1][nt][i] = sgv;
          }
          storeC(hb, WSTR, nt, h, lr, half);
        }
        wfence();
      }
      // delta4 = wo .* sigmoid(z4)
#pragma unroll
      for (int nt = 0; nt < 2; ++nt) {
        float w = smem[O_UWO + nt * 16 + lr];
        v8f d4;
#pragma unroll
        for (int i = 0; i < 8; ++i) d4[i] = w * sg[4][nt][i];
        storeC(gb, WSTR, nt, d4, lr, half);
      }
      wfence();
#pragma unroll
      for (int l = 3; l >= 0; --l) {
        v8f acc[2];
        acc[0] = splat8(0.f);
        acc[1] = splat8(0.f);
        // delta_l = (delta_{l+1} @ Wh[l]^T) .* sigmoid(z_l)
        mmacc<8, 2>(gb, WSTR, smem + O_UPW + l * 32 * WSTR, WSTR, acc, lr, half);
#pragma unroll
        for (int nt = 0; nt < 2; ++nt) {
          v8f dn;
#pragma unroll
          for (int i = 0; i < 8; ++i) dn[i] = acc[nt][i] * sg[l][nt][i];
          storeC(gb, WSTR, nt, dn, lr, half);
        }
        wfence();
      }
      // tau_pot = delta_0 @ W0^T   (N padded to 16, rows>=6 are zero)
      v8f tp[1];
      tp[0] = splat8(0.f);
      mmacc<8, 1>(gb, WSTR, smem + O_UW0P, WSTR, tp, lr, half);
      storeC(pb, 16, 0, tp[0], lr, half);
      wfence();
    }

    // ========= M-MLP forward + JVP (tangent = xd) -> M, dM =========
    {
      const float* Bp = smem;
      {
        v8f pz[2], tz[2];
        pz[0] = splat8(Bp[O_B0 + lr]);
        pz[1] = splat8(Bp[O_B0 + 16 + lr]);
        tz[0] = splat8(0.f);
        tz[1] = splat8(0.f);
        mmacc_dual<2, 2>(xb, tb, 8, Bp + O_WT0, 8, pz, tz, lr, half);
#pragma unroll
        for (int nt = 0; nt < 2; ++nt) {
          v8f h, t;
#pragma unroll
          for (int i = 0; i < 8; ++i) {
            float sp, sgv;
            spsig(pz[nt][i], sp, sgv);
            h[i] = sp;
            t[i] = sgv * tz[nt][i];
          }
          storeC(hb, WSTR, nt, h, lr, half);
          storeC(gb, WSTR, nt, t, lr, half);
        }
        wfence();
      }
#pragma unroll
      for (int l = 0; l < 4; ++l) {
        v8f pz[2], tz[2];
        pz[0] = splat8(Bp[O_BH + l * 32 + lr]);
        pz[1] = splat8(Bp[O_BH + l * 32 + 16 + lr]);
        tz[0] = splat8(0.f);
        tz[1] = splat8(0.f);
        mmacc_dual<8, 2>(hb, gb, WSTR, Bp + O_WTH + l * 32 * WSTR, WSTR, pz, tz,
                         lr, half);
#pragma unroll
        for (int nt = 0; nt < 2; ++nt) {
          v8f h, t;
#pragma unroll
          for (int i = 0; i < 8; ++i) {
            float sp, sgv;
            spsig(pz[nt][i], sp, sgv);
            h[i] = sp;
            t[i] = sgv * tz[nt][i];
          }
          storeC(hb, WSTR, nt, h, lr, half);
          storeC(gb, WSTR, nt, t, lr, half);
        }
        wfence();
      }
      v8f po[3], to[3];
#pragma unroll
      for (int nt = 0; nt < 3; ++nt) {
        po[nt] = splat8(Bp[O_BO + nt * 16 + lr]);
        to[nt] = splat8(0.f);
      }
      mmacc_dual<8, 3>(hb, gb, WSTR, Bp + O_WTO, WSTR, po, to, lr, half);
#pragma unroll
      for (int nt = 0; nt < 3; ++nt) {
        storeC(o1, 48, nt, po[nt], lr, half);
        storeC(o2, 48, nt, to[nt], lr, half);
      }
      wfence();
    }

    // ============ per-sample finalize: rhs assembly + 6x6 solve ============
    if (half == 0) {
      const float* r1 = o1 + lr * 48;
      const float* r2 = o2 + lr * 48;
      float xdv[6], tav[6];
#pragma unroll
      for (int j = 0; j < 6; ++j) {
        xdv[j] = tb[lr * 8 + j];
        tav[j] = taub[lr * 8 + j];
      }
      float Mm[6][7];
#pragma unroll
      for (int i = 0; i < 6; ++i) {
        float tc = 0.f, te = 0.f;
#pragma unroll
        for (int j = 0; j < 6; ++j) {
          tc += 0.5f * (r2[i * 6 + j] + r2[j * 6 + i]) * xdv[j];
          te += tav[j] * smem[O_AM + j * 6 + i];
          Mm[i][j] = 0.5f * (r1[i * 6 + j] + r1[j * 6 + i]);
        }
        Mm[i][6] = te - tc - pb[lr * 16 + i] - td[i];
      }
      // Gauss-Jordan with partial pivoting
      for (int c = 0; c < 6; ++c) {
        int p = c;
        float bv = fabsf(Mm[c][c]);
        for (int r = c + 1; r < 6; ++r) {
          float v = fabsf(Mm[r][c]);
          if (v > bv) { bv = v; p = r; }
        }
        if (p != c) {
          for (int j = 0; j < 7; ++j) {
            float t = Mm[c][j];
            Mm[c][j] = Mm[p][j];
            Mm[p][j] = t;
          }
        }
        float ip = 1.f / Mm[c][c];
        for (int r = 0; r < 6; ++r) {
          if (r == c) continue;
          float f = Mm[r][c] * ip;
          for (int j = 0; j < 7; ++j) Mm[r][j] -= f * Mm[c][j];
        }
      }
      float* orow = outp + (size_t)(s0 + lr) * 6;
#pragma unroll
      for (int i = 0; i < 6; ++i) orow[i] = Mm[i][6] / Mm[i][i];
    }
    wfence();
  }
}

extern "C" void kernel_launch(void* const* d_in, const int* in_sizes, int n_in,
                              void* d_out, int out_size, void* d_ws,
                              size_t ws_size, hipStream_t stream) {
  (void)n_in; (void)out_size; (void)d_ws; (void)ws_size;
  const float* inp = (const float*)d_in[0];
  const float* Am  = (const float*)d_in[1];
  const float* mW0 = (const float*)d_in[2];
  const float* mb0 = (const float*)d_in[3];
  const float* mWh = (const float*)d_in[4];
  const float* mbh = (const float*)d_in[5];
  const float* mWo = (const float*)d_in[6];
  const float* mbo = (const float*)d_in[7];
  const float* uW0 = (const float*)d_in[8];
  const float* ub0 = (const float*)d_in[9];
  const float* uWh = (const float*)d_in[10];
  const float* ubh = (const float*)d_in[11];
  const float* uWo = (const float*)d_in[12];
  const float* ubo = (const float*)d_in[13];
  const float* dW0 = (const float*)d_in[14];
  const float* db0 = (const float*)d_in[15];
  const float* dWh = (const float*)d_in[16];
  const float* dbh = (const float*)d_in[17];
  const float* dWo = (const float*)d_in[18];
  const float* dbo = (const float*)d_in[19];

  int nB = in_sizes[0] / 18;
  int nTiles = nB / 16;

  (void)hipFuncSetAttribute((const void*)lnn_kernel,
                            hipFuncAttributeMaxDynamicSharedMemorySize,
                            SMEM_BYTES);

  int blocks = 256;
  if (blocks * WAVES > nTiles) blocks = (nTiles + WAVES - 1) / WAVES;
  if (blocks < 1) blocks = 1;

  lnn_kernel<<<blocks, BLOCK, SMEM_BYTES, stream>>>(
      inp, Am, mW0, mb0, mWh, mbh, mWo, mbo, uW0, ub0, uWh, ubh, uWo, ubo, dW0,
      db0, dWh, dbh, dWo, dbo, (float*)d_out, nTiles);
}